// EdgeGATConvDiy_80161269613280
// MI455X (gfx1250) — compile-verified
//
#include <hip/hip_runtime.h>
#include <hip/hip_bf16.h>
#include <math.h>

typedef float v2f __attribute__((ext_vector_type(2)));
typedef float v8f __attribute__((ext_vector_type(8)));

#define D  64
#define HP 6   // HOUR_PERIOD

__device__ __forceinline__ float wave_sum32(float v) {
#pragma unroll
  for (int off = 16; off > 0; off >>= 1) v += __shfl_xor(v, off, 32);
  return v;
}

__global__ void k_zero(float* __restrict__ p, int n) {
  int i = blockIdx.x * blockDim.x + threadIdx.x;
  if (i < n) p[i] = 0.0f;
}

// ---------------------------------------------------------------------------
// time_rel table: hour_id has only HP=6 distinct values, so the reference's
// [E,64]x[64,64] matmul collapses to a 6x64 . 64x64 projection done once.
// One wave, V_WMMA_F32_16X16X4_F32 tiles: Y[16x64] = mix[16x64] * W^T[64x64].
// ---------------------------------------------------------------------------
__global__ void k_time_rel(const float* __restrict__ hour_emb,  // [25][D]
                           const float* __restrict__ W,         // [D][D] row-major
                           const float* __restrict__ bias,      // [D]
                           float* __restrict__ trel_tab)        // [HP][D]
{
  __shared__ float mixLds[16 * D];
  __shared__ float yLds[16 * D];
  const int lane = threadIdx.x;  // 0..31, single wave

  // Build mix matrix: rows 0..5 = (hour_emb[last]+hour_emb[h]+hour_emb[next])/3
  for (int idx = lane; idx < 16 * D; idx += 32) {
    int r = idx / D, c = idx % D;
    float v = 0.0f;
    if (r < HP) {
      int lastH = (r + HP - 1) % HP, nextH = (r + 1) % HP;
      v = (hour_emb[lastH * D + c] + hour_emb[r * D + c] + hour_emb[nextH * D + c]) * (1.0f / 3.0f);
    }
    mixLds[idx] = v;
  }
  __syncthreads();

  const int lo = lane & 15, hi = lane >> 4;
#pragma unroll
  for (int jt = 0; jt < 4; ++jt) {     // 4 N-tiles of 16
    v8f acc = {};
    for (int k0 = 0; k0 < D; k0 += 4) {
      // A 16x4 f32 layout: lane l -> M=l%16, K = k0 + v + 2*(l/16)
      v2f a, b;
      a.x = mixLds[lo * D + k0 + 2 * hi];
      a.y = mixLds[lo * D + k0 + 2 * hi + 1];
      // B[k][n] = W[jt*16+n][k]  (Y = mix * W^T)
      b.x = W[(jt * 16 + lo) * D + k0 + 2 * hi];
      b.y = W[(jt * 16 + lo) * D + k0 + 2 * hi + 1];
      acc = __builtin_amdgcn_wmma_f32_16x16x4_f32(false, a, false, b,
                                                  (short)0, acc, false, false);
    }
    // C/D 16x16 f32 layout: lane l holds row v+8*(l/16), col l%16
#pragma unroll
    for (int v = 0; v < 8; ++v) {
      int row = v + 8 * hi;
      yLds[row * D + jt * 16 + lo] = acc[v];
    }
  }
  __syncthreads();

  // bias + L2 normalize (eps clamp like torch F.normalize); lane owns 2 cols
  int c0 = 2 * lane, c1 = 2 * lane + 1;
  for (int r = 0; r < HP; ++r) {
    float v0 = yLds[r * D + c0] + bias[c0];
    float v1 = yLds[r * D + c1] + bias[c1];
    float ss = wave_sum32(v0 * v0 + v1 * v1);
    float inv = 1.0f / fmaxf(sqrtf(ss), 1e-12f);
    trel_tab[r * D + c0] = v0 * inv;
    trel_tab[r * D + c1] = v1 * inv;
  }
}

// ---------------------------------------------------------------------------
// Category-graph edge score (TransE): wave32 per edge, lane owns 2 components.
// ---------------------------------------------------------------------------
__global__ void k_cat_score(const float* __restrict__ cat_emb,
                            const float* __restrict__ rel_emb,
                            const int* __restrict__ csrc, const int* __restrict__ cdst,
                            float* __restrict__ cscore, float* __restrict__ cat_max, int EC)
{
  int g = blockIdx.x * blockDim.x + threadIdx.x;
  int e = g >> 5, lane = g & 31;
  if (e >= EC) return;
  int s = csrc[e], d = cdst[e];
  const float2* ce2 = (const float2*)cat_emb;
  float2 h = ce2[s * 32 + lane];
  float2 t = ce2[d * 32 + lane];
  const float2* r2 = (const float2*)(rel_emb + 6 * D);
  float2 r = r2[lane];
  float dx = h.x + r.x - t.x;
  float dy = h.y + r.y - t.y;
  float ss = wave_sum32(dx * dx + dy * dy);
  if (lane == 0) {
    float sc = expf(-ss);               // > 0 -> int-punned atomicMax is valid
    cscore[e] = sc;
    atomicMax((int*)(cat_max + d), __float_as_int(sc));
  }
}

// ---------------------------------------------------------------------------
// Main-graph edge score (TransH on relation + time hyperplanes).
// d1 = (h-t+rel) + ((t.n)-(h.n)) n ; d2 = (h-t+trel) + ((t.tn)-(h.tn)) tn
// fin = exp(-(|d1|^2 + |d2|^2))
// ---------------------------------------------------------------------------
__global__ void k_edge_score(const float* __restrict__ nfeat,
                             const float* __restrict__ rel_emb,
                             const float* __restrict__ norm_emb,
                             const float* __restrict__ trel_tab,
                             const int* __restrict__ src, const int* __restrict__ dst,
                             const int* __restrict__ ftype, const int* __restrict__ hour,
                             float* __restrict__ escore, float* __restrict__ node_max, int E)
{
  int g = blockIdx.x * blockDim.x + threadIdx.x;
  int e = g >> 5, lane = g & 31;
  if (e >= E) return;
  int s = src[e], d = dst[e], ft = ftype[e], hr = hour[e];
  const float2* nf2 = (const float2*)nfeat;
  float2 h = nf2[s * 32 + lane];
  float2 t = nf2[d * 32 + lane];
  float2 r  = ((const float2*)(rel_emb  + ft * D))[lane];
  float2 n  = ((const float2*)(norm_emb + ft * D))[lane];
  float2 tn = ((const float2*)(norm_emb + 5 * D))[lane];
  float2 tr = ((const float2*)(trel_tab + hr * D))[lane];

  float hn  = wave_sum32(h.x * n.x  + h.y * n.y);
  float tnd = wave_sum32(t.x * n.x  + t.y * n.y);
  float htn = wave_sum32(h.x * tn.x + h.y * tn.y);
  float ttn = wave_sum32(t.x * tn.x + t.y * tn.y);
  float c1 = tnd - hn, c2 = ttn - htn;

  float u0 = h.x - t.x, u1 = h.y - t.y;
  float d10 = u0 + r.x  + c1 * n.x,  d11 = u1 + r.y  + c1 * n.y;
  float d20 = u0 + tr.x + c2 * tn.x, d21 = u1 + tr.y + c2 * tn.y;
  float ss = wave_sum32(d10 * d10 + d11 * d11 + d20 * d20 + d21 * d21);

  if (lane == 0) {
    float fin = expf(-ss);
    escore[e] = fin;
    atomicMax((int*)(node_max + d), __float_as_int(fin));
  }
}

// segment exp-sum (one thread per edge; m stored as float bits from int max)
__global__ void k_sum(const float* __restrict__ escore, const int* __restrict__ dst,
                      const float* __restrict__ seg_max, float* __restrict__ seg_sum, int E)
{
  int e = blockIdx.x * blockDim.x + threadIdx.x;
  if (e >= E) return;
  int d = dst[e];
  atomicAdd(seg_sum + d, expf(escore[e] - seg_max[d]));
}

// weighted scatter: out[dst] += feat[src] * softmax_coeff  (wave per edge)
__global__ void k_aggregate(const float* __restrict__ feat,
                            const float* __restrict__ escore,
                            const int* __restrict__ src, const int* __restrict__ dst,
                            const float* __restrict__ seg_max, const float* __restrict__ seg_sum,
                            float* __restrict__ out, int E)
{
  int g = blockIdx.x * blockDim.x + threadIdx.x;
  int e = g >> 5, lane = g & 31;
  if (e >= E) return;
  int s = src[e], d = dst[e];
  float a = expf(escore[e] - seg_max[d]) / seg_sum[d];
  float2 h = ((const float2*)feat)[s * 32 + lane];
  atomicAdd(out + d * D + 2 * lane,     h.x * a);
  atomicAdd(out + d * D + 2 * lane + 1, h.y * a);
}

extern "C" void kernel_launch(void* const* d_in, const int* in_sizes, int n_in,
                              void* d_out, int out_size, void* d_ws, size_t ws_size,
                              hipStream_t stream) {
  const float* nfeat    = (const float*)d_in[0];
  const float* cat_emb  = (const float*)d_in[1];
  const float* rel_emb  = (const float*)d_in[2];
  const float* norm_emb = (const float*)d_in[3];
  const float* hour_emb = (const float*)d_in[4];
  const float* W        = (const float*)d_in[5];
  const float* bias     = (const float*)d_in[6];
  const int* src   = (const int*)d_in[7];
  const int* dst   = (const int*)d_in[8];
  const int* ftype = (const int*)d_in[9];
  const int* hour  = (const int*)d_in[10];
  const int* csrc  = (const int*)d_in[11];
  const int* cdst  = (const int*)d_in[12];

  const int N  = in_sizes[0] / D;
  const int NC = in_sizes[1] / D;
  const int E  = in_sizes[7];
  const int EC = in_sizes[11];

  // workspace carve (floats): maxes/sums, time_rel table, edge scores
  float* ws        = (float*)d_ws;
  float* node_max  = ws;                       // N
  float* node_sum  = node_max + N;             // N
  float* cat_max   = node_sum + N;             // NC
  float* cat_sum   = cat_max + NC;             // NC
  float* trel_tab  = cat_sum + NC;             // HP*D
  float* escore    = trel_tab + HP * D;        // E
  float* cscore    = escore + E;               // EC

  float* out     = (float*)d_out;              // [N*D] rst
  float* cat_out = out + (size_t)N * D;        // [NC*D] cat_hid

  // init: zero outputs and all softmax stats (0.0f == int 0 works as the max
  // identity because every score is exp(-x) >= 0)
  {
    int zn = N * D + NC * D;
    k_zero<<<(zn + 255) / 256, 256, 0, stream>>>(out, zn);
    int zs = 2 * N + 2 * NC;
    k_zero<<<(zs + 255) / 256, 256, 0, stream>>>(node_max, zs);
  }

  // hoisted time-relation projection via WMMA (one wave)
  k_time_rel<<<1, 32, 0, stream>>>(hour_emb, W, bias, trel_tab);

  // ---- category graph: score/max -> exp-sum -> weighted scatter
  {
    int blocks = (EC * 32 + 255) / 256;
    k_cat_score<<<blocks, 256, 0, stream>>>(cat_emb, rel_emb, csrc, cdst, cscore, cat_max, EC);
    k_sum<<<(EC + 255) / 256, 256, 0, stream>>>(cscore, cdst, cat_max, cat_sum, EC);
    k_aggregate<<<blocks, 256, 0, stream>>>(cat_emb, cscore, csrc, cdst, cat_max, cat_sum, cat_out, EC);
  }

  // ---- main graph: score/max -> exp-sum -> weighted scatter
  {
    int blocks = (int)(((long long)E * 32 + 255) / 256);
    k_edge_score<<<blocks, 256, 0, stream>>>(nfeat, rel_emb, norm_emb, trel_tab,
                                             src, dst, ftype, hour, escore, node_max, E);
    k_sum<<<(E + 255) / 256, 256, 0, stream>>>(escore, dst, node_max, node_sum, E);
    k_aggregate<<<blocks, 256, 0, stream>>>(nfeat, escore, src, dst, node_max, node_sum, out, E);
  }
}